// Model_15324443312668
// MI455X (gfx1250) — compile-verified
//
#include <hip/hip_runtime.h>
#include <stdint.h>

// ---------------------------------------------------------------------------
// Rigid-frame composition kernel for MI455X (gfx1250).
// Memory-bound problem (~3 FLOP/byte, ~228 MB total traffic @ 23.3 TB/s HBM):
// no WMMA (3x3 algebra + sequential data-dependent chain), instead use the
// CDNA5 data-movement path: TDM tensor_load_to_lds for the per-residue-type
// tables, global_prefetch_b8 for the streaming rows, s_wait_tensorcnt, and
// bank-conflict-free LDS scratch for the dynamically indexed frames.
// ---------------------------------------------------------------------------

#define NRES 21
#define MAXR 8
#define MAXT 7
#define MAXA 24
#define TB   128

typedef unsigned int u32x4 __attribute__((ext_vector_type(4)));
typedef int          i32x8 __attribute__((ext_vector_type(8)));
typedef int          i32x4 __attribute__((ext_vector_type(4)));

// Low 32 bits of a generic pointer into LDS == LDS byte offset (aperture is in
// the high 32 bits on gfx1250; CDNA5 ISA).
__device__ __forceinline__ unsigned lds_off(const void* p) {
  return (unsigned)(uintptr_t)p;
}

// 1-D contiguous TDM copy: global -> LDS, nelem f32 elements.
// D# built per CDNA5 ISA:
//   group0: count=1 | lds_addr | global_addr[56:0] | type=2
//   group1: data_size=4B, tensor_dim0=tile_dim0=nelem, tensor_dim1=tile_dim1=1
// remaining groups zero (<=2-D tensor, no gather/iterate). Tracked by
// TENSORcnt. This toolchain carries the 6-arg clang-23 builtin form.
__device__ __forceinline__ void tdm_load_1d(const void* gsrc, unsigned ldsDst,
                                            unsigned nelem) {
  unsigned long long ga = (unsigned long long)(uintptr_t)gsrc;
  u32x4 g0;
  g0.x = 1u;                                                // count=1 (valid)
  g0.y = ldsDst;                                            // lds_addr
  g0.z = (unsigned)ga;                                      // global_addr lo
  g0.w = (unsigned)((ga >> 32) & 0x01FFFFFFull) | (2u << 30); // hi | type=2
  i32x8 g1;
  g1[0] = (int)(2u << 16);                       // data_size = 4 bytes
  g1[1] = (int)((nelem & 0xFFFFu) << 16);        // tensor_dim0[15:0]
  g1[2] = (int)((nelem >> 16) | (1u << 16));     // tensor_dim0 hi | tensor_dim1=1
  g1[3] = (int)((nelem & 0xFFFFu) << 16);        // tile_dim0 = nelem (<65536)
  g1[4] = 1;                                     // tile_dim1=1, tile_dim2=0
  g1[5] = (int)nelem;                            // tensor_dim0_stride lo
  g1[6] = 0;
  g1[7] = 0;
  i32x4 z4; z4[0] = 0; z4[1] = 0; z4[2] = 0; z4[3] = 0;
  i32x8 z8; z8[0] = 0; z8[1] = 0; z8[2] = 0; z8[3] = 0;
  z8[4] = 0; z8[5] = 0; z8[6] = 0; z8[7] = 0;
  __builtin_amdgcn_tensor_load_to_lds(g0, g1, z4, z4, z8, 0);
}

// Z = X o Y on [4][3] frames (rows 0..2 = R row-major, row 3 = t):
//   Z.R = X.R @ Y.R ; Z.t = X.R @ Y.t + X.t
__device__ __forceinline__ void combine_full(const float* __restrict__ X,
                                             const float* __restrict__ Y,
                                             float* __restrict__ Z) {
#pragma unroll
  for (int i = 0; i < 3; ++i) {
    const float x0 = X[i * 3 + 0], x1 = X[i * 3 + 1], x2 = X[i * 3 + 2];
#pragma unroll
    for (int k = 0; k < 3; ++k)
      Z[i * 3 + k] = fmaf(x0, Y[k], fmaf(x1, Y[3 + k], x2 * Y[6 + k]));
    Z[9 + i] = fmaf(x0, Y[9], fmaf(x1, Y[10], fmaf(x2, Y[11], X[9 + i])));
  }
}

__global__ __launch_bounds__(TB) void frames_kernel(
    const int* __restrict__ rtype,       // (N)
    const float* __restrict__ pos,       // (N,3)
    const float* __restrict__ bb,        // (N,4,3)
    const float* __restrict__ sc,        // (N,7,2)
    const float* __restrict__ transforms,// (21,8,4,3)
    const float* __restrict__ rigids,    // (21,24,3)
    const int* __restrict__ tdep,        // (21,8)
    const int* __restrict__ gdep,        // (21,24)
    float* __restrict__ out,             // N*72 (R) then N*12 (frame0)
    int n_res) {
  __shared__ float sT[NRES * MAXR * 12];       // 8064 B
  __shared__ float sG[NRES * MAXA * 3];        // 6048 B
  __shared__ float sF[MAXR * 12 * TB];         // 49152 B, [frame][comp][lane]

  const int t = threadIdx.x;
  const int n = blockIdx.x * TB + t;

  // Stage constant tables via the Tensor Data Mover (one wave issues; TDM
  // ignores EXEC; TENSORcnt is per-wave so only the issuing wave waits >0).
  if (t == 0) {
    tdm_load_1d(transforms, lds_off(sT), NRES * MAXR * 12);
    tdm_load_1d(rigids, lds_off(sG), NRES * MAXA * 3);
  }
  // Overlap: pull this thread's streaming rows toward the caches while the
  // TDM runs (global_prefetch_b8 fetches the whole cacheline).
  if (n < n_res) {
    __builtin_prefetch(bb + (size_t)n * 12, 0, 3);
    __builtin_prefetch(sc + (size_t)n * MAXT * 2, 0, 3);
  }
  __builtin_amdgcn_s_wait_tensorcnt(0);
  __syncthreads();

  if (n >= n_res) return;

  const int rt = rtype[n];

  // -------- frame 0: backbone, then compose with T[rt][0] --------
  {
    float Y[12];
    const float4* b4 = (const float4*)(bb + (size_t)n * 12);  // 48B-aligned
    const float4 a = b4[0], b = b4[1], c = b4[2];
    Y[0] = a.x; Y[1] = a.y; Y[2] = a.z; Y[3] = a.w;
    Y[4] = b.x; Y[5] = b.y; Y[6] = b.z; Y[7] = b.w;
    Y[8] = c.x;
    const float* pp = pos + (size_t)n * 3;
    Y[9] = c.y + pp[0]; Y[10] = c.z + pp[1]; Y[11] = c.w + pp[2];

    float Z[12];
    combine_full(&sT[(rt * MAXR + 0) * 12], Y, Z);
#pragma unroll
    for (int k = 0; k < 12; ++k) sF[(0 * 12 + k) * TB + t] = Z[k];
  }

  // -------- frames 1..7: T[rt][i] o Rx(c,s); Rx = [[1,0,0],[0,c,-s],[0,s,c]]
  // Specialized compose: R'[:,0]=X[:,0]; R'[:,1]=X[:,1]c+X[:,2]s;
  //                      R'[:,2]=X[:,2]c-X[:,1]s; t'=X.t
  const float2* s2 = (const float2*)(sc + (size_t)n * MAXT * 2);  // 8B-aligned
#pragma unroll
  for (int i = 1; i < MAXR; ++i) {
    const float2 cs = s2[i - 1];
    const float cc = cs.x, ss = cs.y;
    const float* X = &sT[(rt * MAXR + i) * 12];
    float Z[12];
#pragma unroll
    for (int r = 0; r < 3; ++r) {
      const float x0 = X[r * 3 + 0], x1 = X[r * 3 + 1], x2 = X[r * 3 + 2];
      Z[r * 3 + 0] = x0;
      Z[r * 3 + 1] = fmaf(x1, cc, x2 * ss);
      Z[r * 3 + 2] = fmaf(x2, cc, -x1 * ss);
      Z[9 + r] = X[9 + r];
    }
#pragma unroll
    for (int k = 0; k < 12; ++k) sF[(i * 12 + k) * TB + t] = Z[k];
  }

  // -------- sequential dependency chain (dynamic frame index -> LDS) -------
  int td[MAXR];
  {
    const int4* tp = (const int4*)(tdep + rt * MAXR);  // 32B-aligned
    const int4 t0 = tp[0], t1 = tp[1];
    td[0] = t0.x; td[1] = t0.y; td[2] = t0.z; td[3] = t0.w;
    td[4] = t1.x; td[5] = t1.y; td[6] = t1.z; td[7] = t1.w;
  }
#pragma unroll
  for (int i = 1; i < MAXR; ++i) {
    const int d = td[i];
    float P[12], C[12], Z[12];
#pragma unroll
    for (int k = 0; k < 12; ++k) P[k] = sF[(d * 12 + k) * TB + t];
#pragma unroll
    for (int k = 0; k < 12; ++k) C[k] = sF[(i * 12 + k) * TB + t];
    combine_full(P, C, Z);
#pragma unroll
    for (int k = 0; k < 12; ++k) sF[(i * 12 + k) * TB + t] = Z[k];
  }

  // -------- 24 atom placements: R[a] = F[gd[a]].R @ G[a] + F[gd[a]].t ------
  int gd[MAXA];
  {
    const int4* gp = (const int4*)(gdep + rt * MAXA);  // 96B offset, 16B-aligned
#pragma unroll
    for (int q = 0; q < 6; ++q) {
      const int4 v = gp[q];
      gd[q * 4 + 0] = v.x; gd[q * 4 + 1] = v.y;
      gd[q * 4 + 2] = v.z; gd[q * 4 + 3] = v.w;
    }
  }
  float* outR = out + (size_t)n * 72;  // 288B/thread contiguous, 16B-aligned
#pragma unroll
  for (int a0 = 0; a0 < MAXA; a0 += 4) {
    float tmp[12];
#pragma unroll
    for (int a = 0; a < 4; ++a) {
      const int d = gd[a0 + a];
      float F[12];
#pragma unroll
      for (int k = 0; k < 12; ++k) F[k] = sF[(d * 12 + k) * TB + t];
      const float* g = &sG[(rt * MAXA + a0 + a) * 3];
      const float gx = g[0], gy = g[1], gz = g[2];
      tmp[a * 3 + 0] = fmaf(F[0], gx, fmaf(F[1], gy, fmaf(F[2], gz, F[9])));
      tmp[a * 3 + 1] = fmaf(F[3], gx, fmaf(F[4], gy, fmaf(F[5], gz, F[10])));
      tmp[a * 3 + 2] = fmaf(F[6], gx, fmaf(F[7], gy, fmaf(F[8], gz, F[11])));
    }
    float4* o4 = (float4*)(outR + a0 * 3);
    o4[0] = make_float4(tmp[0], tmp[1], tmp[2], tmp[3]);
    o4[1] = make_float4(tmp[4], tmp[5], tmp[6], tmp[7]);
    o4[2] = make_float4(tmp[8], tmp[9], tmp[10], tmp[11]);
  }

  // -------- second output: final frame selected by gdep[rt][0] -------------
  {
    const int d = gd[0];
    float F[12];
#pragma unroll
    for (int k = 0; k < 12; ++k) F[k] = sF[(d * 12 + k) * TB + t];
    float4* o4 = (float4*)(out + (size_t)n_res * 72 + (size_t)n * 12);
    o4[0] = make_float4(F[0], F[1], F[2], F[3]);
    o4[1] = make_float4(F[4], F[5], F[6], F[7]);
    o4[2] = make_float4(F[8], F[9], F[10], F[11]);
  }
}

extern "C" void kernel_launch(void* const* d_in, const int* in_sizes, int n_in,
                              void* d_out, int out_size, void* d_ws,
                              size_t ws_size, hipStream_t stream) {
  const int* rtype      = (const int*)d_in[0];
  const float* pos      = (const float*)d_in[1];
  const float* bb       = (const float*)d_in[2];
  const float* sc       = (const float*)d_in[3];
  const float* transforms = (const float*)d_in[4];
  const float* rigids   = (const float*)d_in[5];
  const int* tdep       = (const int*)d_in[6];
  const int* gdep       = (const int*)d_in[7];
  float* out            = (float*)d_out;
  const int n = in_sizes[0];

  dim3 grid((n + TB - 1) / TB), block(TB);
  hipLaunchKernelGGL(frames_kernel, grid, block, 0, stream, rtype, pos, bb, sc,
                     transforms, rigids, tdep, gdep, out, n);
}